// GIN_72043781423169
// MI455X (gfx1250) — compile-verified
//
#include <hip/hip_runtime.h>

#define N_NODES 100000
#define N_EDGES 1600000
#define DFEAT   128

typedef __attribute__((ext_vector_type(2))) float v2f;
typedef __attribute__((ext_vector_type(8))) float v8f;

// ---------------- zero workspace ----------------
__global__ __launch_bounds__(256) void gin_zero_kernel(float4* __restrict__ p, int n4) {
    int i = blockIdx.x * 256 + threadIdx.x;
    if (i < n4) p[i] = make_float4(0.f, 0.f, 0.f, 0.f);
}

// ---------------- edge scatter: agg[dst] += feat[src] ----------------
// One wave32 handles one edge: lane L covers features [4L, 4L+4).
// Loads are fully coalesced (512B per wave from one source row); the 51.2MB
// agg buffer lives in L2 (192MB), so the f32 atomics resolve on-chip.
__global__ __launch_bounds__(256) void gin_scatter_kernel(const float* __restrict__ feat,
                                                          const int* __restrict__ src,
                                                          const int* __restrict__ dst,
                                                          float* __restrict__ agg) {
    int idx = blockIdx.x * 256 + threadIdx.x;          // < N_EDGES * 32
    int e = idx >> 5;
    if (e >= N_EDGES) return;
    int c = (idx & 31) << 2;                           // feature column (0..124 step 4)
    int s = src[e];
    int d = dst[e];
    const float4 v = *(const float4*)(feat + (long long)s * DFEAT + c);
    float* ap = agg + (long long)d * DFEAT + c;
    atomicAdd(ap + 0, v.x);
    atomicAdd(ap + 1, v.y);
    atomicAdd(ap + 2, v.z);
    atomicAdd(ap + 3, v.w);
}

// ---------------- fused (xin + agg) @ W^T + b via V_WMMA_F32_16X16X4_F32 ----
// One wave per block (EXEC all-1s as WMMA requires). Each wave computes a
// 16-row x 128-col output block: 8 column tiles, K swept in steps of 4.
// A fragment (x+agg) is reused across all 8 column-tile WMMAs per K-step.
__global__ __launch_bounds__(32) void gin_gemm_kernel(const float* __restrict__ xin,
                                                      const float* __restrict__ agg,
                                                      const float* __restrict__ W,   // [out=128][in=128]
                                                      const float* __restrict__ bias,
                                                      float* __restrict__ out) {
    const int lane = threadIdx.x;        // 0..31
    const int half = lane >> 4;          // 0: lower half-lanes, 1: upper
    const int l15  = lane & 15;

    v8f acc[8];
    v8f zero = {};
#pragma unroll
    for (int t = 0; t < 8; ++t) acc[t] = zero;

    const long long rowA = (long long)blockIdx.x * 16 + l15;   // A-matrix row for this lane
    const float* __restrict__ xrow = xin + rowA * DFEAT;
    const float* __restrict__ arow = agg + rowA * DFEAT;

#pragma unroll 4
    for (int k0 = 0; k0 < DFEAT; k0 += 4) {
        const int kA = k0 + half * 2;    // K pair this lane supplies: {kA, kA+1}
        const float2 xv = *(const float2*)(xrow + kA);
        const float2 av = *(const float2*)(arow + kA);
        v2f a;
        a.x = xv.x + av.x;
        a.y = xv.y + av.y;
#pragma unroll
        for (int nt = 0; nt < 8; ++nt) {
            const int n = nt * 16 + l15; // B column for this lane
            // B[k][n] = W[n][k]; aligned float2 covers {kA, kA+1}
            const float2 wv = *(const float2*)(W + (long long)n * DFEAT + kA);
            v2f b;
            b.x = wv.x;
            b.y = wv.y;
            acc[nt] = __builtin_amdgcn_wmma_f32_16x16x4_f32(
                /*neg_a=*/false, a, /*neg_b=*/false, b,
                /*c_mod=*/(short)0, acc[nt], /*reuse_a=*/false, /*reuse_b=*/false);
        }
    }

    // C/D layout: VGPR v -> M = v + 8*half, N = l15 (+16*nt)
    const long long rowBase = (long long)blockIdx.x * 16 + half * 8;
#pragma unroll
    for (int nt = 0; nt < 8; ++nt) {
        const int n = nt * 16 + l15;
        const float bv = bias[n];
#pragma unroll
        for (int v = 0; v < 8; ++v) {
            out[(rowBase + v) * DFEAT + n] = acc[nt][v] + bv;
        }
    }
}

extern "C" void kernel_launch(void* const* d_in, const int* in_sizes, int n_in,
                              void* d_out, int out_size, void* d_ws, size_t ws_size,
                              hipStream_t stream) {
    const float* x    = (const float*)d_in[0];
    const int*   edge = (const int*)d_in[1];     // [2, N_EDGES] row-major
    const float* W1   = (const float*)d_in[2];
    const float* b1   = (const float*)d_in[3];
    const float* W2   = (const float*)d_in[4];
    const float* b2   = (const float*)d_in[5];

    const int* src = edge;                       // edge_index[0]
    const int* dst = edge + N_EDGES;             // edge_index[1]

    float* out = (float*)d_out;                            // first output  [N, D]
    float* hid = out + (size_t)N_NODES * DFEAT;            // second output [N, D]
    float* agg = (float*)d_ws;                             // scratch       [N, D]

    const int n4 = (N_NODES * DFEAT) / 4;                  // 3.2M float4s
    dim3 zeroGrid((n4 + 255) / 256), blk256(256);
    dim3 scatGrid((N_EDGES * 32 + 255) / 256);             // 200000 blocks
    dim3 gemmGrid(N_NODES / 16), blk32(32);                // 6250 blocks, exact

    // Layer 1: hid = (x + segsum(x[src]->dst)) @ W1^T + b1
    gin_zero_kernel<<<zeroGrid, blk256, 0, stream>>>((float4*)agg, n4);
    gin_scatter_kernel<<<scatGrid, blk256, 0, stream>>>(x, src, dst, agg);
    gin_gemm_kernel<<<gemmGrid, blk32, 0, stream>>>(x, agg, W1, b1, hid);

    // Layer 2: out = (hid + segsum(hid[src]->dst)) @ W2^T + b2
    gin_zero_kernel<<<zeroGrid, blk256, 0, stream>>>((float4*)agg, n4);
    gin_scatter_kernel<<<scatGrid, blk256, 0, stream>>>(hid, src, dst, agg);
    gin_gemm_kernel<<<gemmGrid, blk32, 0, stream>>>(hid, agg, W2, b2, out);
}